// GNN_23914377904291
// MI455X (gfx1250) — compile-verified
//
#include <hip/hip_runtime.h>
#include <math.h>

typedef __attribute__((ext_vector_type(2))) float v2f;
typedef __attribute__((ext_vector_type(8))) float v8f;

#define D 128
#define LDA 132   // padded LDS row stride: (m*132)%64 = 4m -> conflict-free column reads

// ---------------------------------------------------------------------------
// GEMM: Out[M x 128] = A[M x 128] @ W[128 x 128]
//   - A tile (16x128) staged once per block in LDS, shared by all 8 waves
//   - one wave per 16x16 tile via V_WMMA_F32_16X16X4_F32 (full fp32)
//   - epilogue modes:
//       bias!=null / doRelu=1 : Out = relu(res + bias)           (MLP layers)
//       Out2!=null            : Out = res; Out2 = res*dinv[row]^2 (conv: hw + self-loop agg)
// Block = 256 threads = 8 waves (wave w owns N-tile w). blockIdx.x = M-tile.
// M must be a multiple of 16 (100000 and 512 both are) -> EXEC all-1s at WMMA.
// ---------------------------------------------------------------------------
__global__ __launch_bounds__(256) void gemm128_wmma(
    const float* __restrict__ A, const float* __restrict__ W,
    const float* __restrict__ bias, const float* __restrict__ dinv,
    float* __restrict__ Out, float* __restrict__ Out2, int doRelu)
{
    __shared__ float As[16][LDA];

    const int lane = threadIdx.x & 31;
    const int wave = threadIdx.x >> 5;        // 0..7 -> N tile
    const int row0 = blockIdx.x << 4;         // M tile * 16
    const int m    = lane & 15;               // A row within tile
    const int half = lane >> 4;               // 0: K+0/1, 1: K+2/3
    const int ncol = (wave << 4) + (lane & 15);

    // Cooperative A-tile stage: 256 threads x 8 floats = 16x128
    {
        const int r  = threadIdx.x >> 4;          // 0..15
        const int c0 = (threadIdx.x & 15) << 3;   // 0,8,...,120
        const float* src = A + (size_t)(row0 + r) * D + c0;
        *(float4*)&As[r][c0]     = *(const float4*)(src);
        *(float4*)&As[r][c0 + 4] = *(const float4*)(src + 4);
    }
    __syncthreads();

    v8f acc = {};
    #pragma unroll 4
    for (int kk = 0; kk < D; kk += 4) {
        const int ka = kk + (half << 1);
        v2f a, b;
        a.x = As[m][ka];                          // ds_load_b64 (8B aligned)
        a.y = As[m][ka + 1];
        b.x = W[(size_t)ka * D + ncol];           // 64KB weight: L1/L2 resident
        b.y = W[(size_t)(ka + 1) * D + ncol];
        acc = __builtin_amdgcn_wmma_f32_16x16x4_f32(
            false, a, false, b, (short)0, acc, false, false);
    }

    const float bv = bias ? bias[ncol] : 0.0f;
    #pragma unroll
    for (int r = 0; r < 8; ++r) {
        const int row = row0 + r + (half << 3);   // C/D layout: M = r + 8*(lane/16)
        float v = acc[r] + bv;
        if (doRelu) v = fmaxf(v, 0.0f);
        Out[(size_t)row * D + ncol] = v;
        if (Out2) {                               // fused self-loop message
            const float s = dinv[row];
            Out2[(size_t)row * D + ncol] = v * (s * s);
        }
    }
}

// ---------------------------------------------------------------------------
// Degree / normalization
// ---------------------------------------------------------------------------
__global__ void deg_init(float* deg, int N) {
    int n = blockIdx.x * blockDim.x + threadIdx.x;
    if (n < N) deg[n] = 1.0f;                     // self-loop
}
__global__ void deg_count(const int* __restrict__ dst, float* deg, int E) {
    int e = blockIdx.x * blockDim.x + threadIdx.x;
    if (e < E) atomicAdd(&deg[dst[e]], 1.0f);
}
__global__ void deg_to_dinv(float* deg, int N) {
    int n = blockIdx.x * blockDim.x + threadIdx.x;
    if (n < N) deg[n] = rsqrtf(deg[n]);           // deg >= 1 always
}

// ---------------------------------------------------------------------------
// Edge scatter: one wave per edge, float4 per lane (32*4 = 128 features).
// agg[dst] += hw[src] * dinv[src]*dinv[dst]   via global_atomic_add_f32.
// hw (51MB) and agg (51MB) are L2-resident on the 192MB L2.
// ---------------------------------------------------------------------------
__global__ __launch_bounds__(256) void edge_scatter(
    const float* __restrict__ hw, const int* __restrict__ src,
    const int* __restrict__ dst, const float* __restrict__ dinv,
    float* __restrict__ agg, int E)
{
    const int e = blockIdx.x * 8 + (threadIdx.x >> 5);
    if (e >= E) return;
    const int lane = threadIdx.x & 31;
    const int s = src[e];
    const int d = dst[e];
    const float nrm = dinv[s] * dinv[d];
    const float4 v = ((const float4*)(hw + (size_t)s * D))[lane];
    float* ap = agg + (size_t)d * D + lane * 4;
    atomicAdd(ap + 0, v.x * nrm);
    atomicAdd(ap + 1, v.y * nrm);
    atomicAdd(ap + 2, v.z * nrm);
    atomicAdd(ap + 3, v.w * nrm);
}

// ---------------------------------------------------------------------------
// h = relu(agg + bias)
// ---------------------------------------------------------------------------
__global__ void bias_relu(const float* __restrict__ agg,
                          const float* __restrict__ b,
                          float* __restrict__ h, int total) {
    int i = blockIdx.x * blockDim.x + threadIdx.x;
    if (i < total) h[i] = fmaxf(agg[i] + b[i & (D - 1)], 0.0f);
}

// ---------------------------------------------------------------------------
// Pooling: pooled[batch[n], d] += h[n, d]   (512x128 = L2-trivial atomics)
// ---------------------------------------------------------------------------
__global__ void zero_buf(float* p, int total) {
    int i = blockIdx.x * blockDim.x + threadIdx.x;
    if (i < total) p[i] = 0.0f;
}
__global__ void pool_scatter(const float* __restrict__ h,
                             const int* __restrict__ batch,
                             float* __restrict__ pooled, int total) {
    int i = blockIdx.x * blockDim.x + threadIdx.x;
    if (i >= total) return;
    const int n = i >> 7;
    const int d = i & (D - 1);
    atomicAdd(&pooled[(size_t)batch[n] * D + d], h[i]);
}

// ---------------------------------------------------------------------------
// Head: out[g] = sigmoid(dot(z[g,:], W3) + b3). One wave per graph (wave32).
// ---------------------------------------------------------------------------
__global__ __launch_bounds__(256) void head_sigmoid(
    const float* __restrict__ z, const float* __restrict__ W3,
    const float* __restrict__ b3, float* __restrict__ out, int G)
{
    const int g = blockIdx.x * 8 + (threadIdx.x >> 5);
    if (g >= G) return;
    const int lane = threadIdx.x & 31;
    float s = 0.0f;
    #pragma unroll
    for (int i = lane; i < D; i += 32) s += z[(size_t)g * D + i] * W3[i];
    #pragma unroll
    for (int o = 16; o > 0; o >>= 1) s += __shfl_xor(s, o, 32);
    if (lane == 0) out[g] = 1.0f / (1.0f + __expf(-(s + b3[0])));
}

// ---------------------------------------------------------------------------
extern "C" void kernel_launch(void* const* d_in, const int* in_sizes, int n_in,
                              void* d_out, int out_size, void* d_ws, size_t ws_size,
                              hipStream_t stream) {
    const float* x     = (const float*)d_in[0];
    const int*   ei    = (const int*)d_in[1];
    const int*   batch = (const int*)d_in[2];
    const float* convW = (const float*)d_in[3];
    const float* convB = (const float*)d_in[4];
    const float* W1    = (const float*)d_in[5];
    const float* b1    = (const float*)d_in[6];
    const float* W2    = (const float*)d_in[7];
    const float* b2    = (const float*)d_in[8];
    const float* W3    = (const float*)d_in[9];
    const float* b3    = (const float*)d_in[10];
    float* out = (float*)d_out;

    const int N = in_sizes[0] / D;        // 100000
    const int E = in_sizes[1] / 2;        // 640000
    const int G = out_size;               // 512
    const int total = N * D;              // 12.8M

    const int* src = ei;
    const int* dst = ei + E;

    // Workspace layout (floats)
    float* ws = (float*)d_ws;
    const size_t BUF = (size_t)N * D;
    float* h      = ws;                   // 51.2 MB
    float* hw     = ws + BUF;             // 51.2 MB
    float* agg    = ws + 2 * BUF;         // 51.2 MB
    float* dinv   = ws + 3 * BUF;         // N floats (deg -> dinv in place)
    float* pooled = dinv + (size_t)N;     // 512*128
    float* z1     = pooled + (size_t)G * D;
    float* z2     = z1 + (size_t)G * D;

    const int T = 256;
    const int mtilesN = N / 16;           // 6250
    const int mtilesG = G / 16;           // 32

    // --- normalization ---
    deg_init<<<(N + T - 1) / T, T, 0, stream>>>(dinv, N);
    deg_count<<<(E + T - 1) / T, T, 0, stream>>>(dst, dinv, E);
    deg_to_dinv<<<(N + T - 1) / T, T, 0, stream>>>(dinv, N);

    // --- 4 GCN layers ---
    for (int l = 0; l < 4; ++l) {
        const float* hin = (l == 0) ? x : h;
        // hw = hin @ convW[l]; agg = hw * dinv^2 (fused self-loop init)
        gemm128_wmma<<<mtilesN, T, 0, stream>>>(hin, convW + (size_t)l * D * D,
                                                nullptr, dinv, hw, agg, 0);
        edge_scatter<<<(E + 7) / 8, T, 0, stream>>>(hw, src, dst, dinv, agg, E);
        bias_relu<<<(total + T - 1) / T, T, 0, stream>>>(agg, convB + (size_t)l * D,
                                                         h, total);
    }

    // --- pooling ---
    zero_buf<<<(G * D + T - 1) / T, T, 0, stream>>>(pooled, G * D);
    pool_scatter<<<(total + T - 1) / T, T, 0, stream>>>(h, batch, pooled, total);

    // --- MLP head (WMMA, fused bias+relu) ---
    gemm128_wmma<<<mtilesG, T, 0, stream>>>(pooled, W1, b1, nullptr, z1, nullptr, 1);
    gemm128_wmma<<<mtilesG, T, 0, stream>>>(z1, W2, b2, nullptr, z2, nullptr, 1);
    head_sigmoid<<<(G + 7) / 8, T, 0, stream>>>(z2, W3, b3, out, G);
}